// NeuralODESolver_1649267442193
// MI455X (gfx1250) — compile-verified
//
#include <hip/hip_runtime.h>
#include <math.h>

// ---------------------------------------------------------------------------
// Batched adaptive Tsit5 Neural-ODE integrator for MI455X (gfx1250).
// All matmuls use V_WMMA_F32_16X16X4_F32 (full fp32 matrix pipe) so the
// adaptive step controller (RTOL=ATOL=1.4e-8, i.e. ~fp32 eps) sees bitwise
// fp32 math. Weights live transposed in LDS; the whole 128-step scan runs
// inside one kernel launch (compute-bound: ~64 GFLOP vs <1 MB of HBM traffic).
// ---------------------------------------------------------------------------

typedef __attribute__((ext_vector_type(2))) float v2f;
typedef __attribute__((ext_vector_type(8))) float v8f;

#define N_B       1024
#define STATE     64
#define CTRL      16
#define WIDTH     128
#define ZDIM      80        // STATE + CTRL
#define TILE      16        // batch rows per workgroup
#define NTHREADS  256       // 8 wave32s
#define MAX_STEPS 128

// padded LDS row strides (avoid 64-bank conflicts on A-fragment reads)
#define SZ_LD 81
#define SH_LD 132

// ---- dynamic-LDS layout (float offsets) ----
#define OFF_W0T  0
#define OFF_W1T  (OFF_W0T + ZDIM * WIDTH)        // 10240
#define OFF_W2T  (OFF_W1T + WIDTH * WIDTH)       // 26624
#define OFF_B0   (OFF_W2T + WIDTH * STATE)       // 34816
#define OFF_B1   (OFF_B0 + WIDTH)
#define OFF_B2   (OFF_B1 + WIDTH)
#define OFF_Z    (OFF_B2 + STATE)                // 16 x 81 stage input (y|u)
#define OFF_H1   (OFF_Z + TILE * SZ_LD)          // 16 x 132 hidden 1
#define OFF_H2   (OFF_H1 + TILE * SH_LD)         // 16 x 132 hidden 2
#define OFF_Y    (OFF_H2 + TILE * SH_LD)         // 16 x 64
#define OFF_Y5   (OFF_Y + TILE * STATE)
#define OFF_K    (OFF_Y5 + TILE * STATE)         // 7 slots of 16 x 64
#define OFF_T    (OFF_K + 7 * TILE * STATE)
#define OFF_DT   (OFF_T + TILE)
#define OFF_DTC  (OFF_DT + TILE)
#define OFF_STEP (OFF_DTC + TILE)
#define OFF_DONE (OFF_STEP + TILE)
#define OFF_NST  (OFF_DONE + TILE)               // int32 view
#define SMEM_FLOATS (OFF_NST + TILE)             // ~195 KB < 320 KB/WGP

// ---- Tsit5 coefficients (fp32, matching jnp.float32 math) ----
#define DT0f   (60.0f / 3600.0f)
#define RTOLf  1.4e-08f
#define ATOLf  1.4e-08f
#define SAFETYf 0.9f
#define FMINf  0.1f
#define FMAXf  10.0f
#define EXPf   0.2f
#define A21f  0.161f
#define A31f (-0.008480655492356989f)
#define A32f  0.335480655492357f
#define A41f  2.8971530571054935f
#define A42f (-6.359448489975075f)
#define A43f  4.3622954328695815f
#define A51f  5.325864828439257f
#define A52f (-11.748883564062828f)
#define A53f  7.4955393428898365f
#define A54f (-0.09249506636175525f)
#define A61f  5.86145544294642f
#define A62f (-12.92096931784711f)
#define A63f  8.159367898576159f
#define A64f (-0.071584973281401f)
#define A65f (-0.028269050394068383f)
#define B1f   0.09646076681806523f
#define B2f   0.01f
#define B3f   0.4798896504144996f
#define B4f   1.379008574103742f
#define B5f  (-3.290069515436081f)
#define B6f   2.324710524099774f
#define E1f (-0.001780011052226f)
#define E2f (-0.000816434459657f)
#define E3f   0.007880878010262f
#define E4f (-0.144711007173263f)
#define E5f   0.582357165452555f
#define E6f (-0.458082105929187f)
#define E7f  (-1.0f / 66.0f)

// ---------------------------------------------------------------------------
// One MLP evaluation on the 16-row tile resident in LDS.
//   input : smem[OFF_Z]   (16 x 80, row stride SZ_LD)   [y_arg | u]
//   output: kout          (16 x 64, row stride STATE)
// Wave w produces output columns [16w, 16w+16). Every 16x16 output tile is a
// chain of f32 WMMAs over K. noinline => single code copy (84 wmma sites).
// ---------------------------------------------------------------------------
__device__ __attribute__((noinline)) void mlp_eval(float* __restrict__ smem,
                                                   float* __restrict__ kout,
                                                   int wave, int lane)
{
    const float* __restrict__ W0t = smem + OFF_W0T;  // [k][n] k<80,  n<128
    const float* __restrict__ W1t = smem + OFF_W1T;  // [k][n] k<128, n<128
    const float* __restrict__ W2t = smem + OFF_W2T;  // [k][n] k<128, n<64
    float* __restrict__ z  = smem + OFF_Z;
    float* __restrict__ h1 = smem + OFF_H1;
    float* __restrict__ h2 = smem + OFF_H2;

    const int mrow  = lane & 15;        // A-fragment row / B-fragment column
    const int hi    = (lane >> 4) & 1;  // half-wave select
    const int koff  = hi * 2;           // A: lanes 16-31 carry K+2, K+3
    const int rbase = hi * 8;           // C/D: lanes 16-31 carry rows 8..15

    // ---- layer 1: (16x80) @ W0t -> tanh -> h1 (16x128) ----
    {
        const int col = wave * 16 + mrow;
        const float bias = smem[OFF_B0 + col];   // per-column => per-lane const
        v8f acc;
#pragma unroll
        for (int i = 0; i < 8; ++i) acc[i] = bias;
#pragma unroll
        for (int k0 = 0; k0 < ZDIM; k0 += 4) {
            v2f a, b;
            a.x = z[mrow * SZ_LD + k0 + koff];
            a.y = z[mrow * SZ_LD + k0 + koff + 1];
            b.x = W0t[(k0 + koff) * WIDTH + col];
            b.y = W0t[(k0 + koff + 1) * WIDTH + col];
            acc = __builtin_amdgcn_wmma_f32_16x16x4_f32(
                false, a, false, b, (short)0, acc, false, false);
        }
#pragma unroll
        for (int i = 0; i < 8; ++i)
            h1[(rbase + i) * SH_LD + col] = tanhf(acc[i]);
    }
    __syncthreads();

    // ---- layer 2: h1 (16x128) @ W1t -> tanh -> h2 (16x128) ----
    {
        const int col = wave * 16 + mrow;
        const float bias = smem[OFF_B1 + col];
        v8f acc;
#pragma unroll
        for (int i = 0; i < 8; ++i) acc[i] = bias;
#pragma unroll
        for (int k0 = 0; k0 < WIDTH; k0 += 4) {
            v2f a, b;
            a.x = h1[mrow * SH_LD + k0 + koff];
            a.y = h1[mrow * SH_LD + k0 + koff + 1];
            b.x = W1t[(k0 + koff) * WIDTH + col];
            b.y = W1t[(k0 + koff + 1) * WIDTH + col];
            acc = __builtin_amdgcn_wmma_f32_16x16x4_f32(
                false, a, false, b, (short)0, acc, false, false);
        }
#pragma unroll
        for (int i = 0; i < 8; ++i)
            h2[(rbase + i) * SH_LD + col] = tanhf(acc[i]);
    }
    __syncthreads();

    // ---- layer 3: h2 (16x128) @ W2t -> kout (16x64); waves 0..3 only ----
    // Whole-wave branch keeps EXEC all-ones inside the WMMA chain.
    if (wave < 4) {
        const int col = wave * 16 + mrow;
        const float bias = smem[OFF_B2 + col];
        v8f acc;
#pragma unroll
        for (int i = 0; i < 8; ++i) acc[i] = bias;
#pragma unroll
        for (int k0 = 0; k0 < WIDTH; k0 += 4) {
            v2f a, b;
            a.x = h2[mrow * SH_LD + k0 + koff];
            a.y = h2[mrow * SH_LD + k0 + koff + 1];
            b.x = W2t[(k0 + koff) * STATE + col];
            b.y = W2t[(k0 + koff + 1) * STATE + col];
            acc = __builtin_amdgcn_wmma_f32_16x16x4_f32(
                false, a, false, b, (short)0, acc, false, false);
        }
#pragma unroll
        for (int i = 0; i < 8; ++i)
            kout[(rbase + i) * STATE + col] = acc[i];
    }
    __syncthreads();
}

// ---------------------------------------------------------------------------
__global__ void __launch_bounds__(NTHREADS)
ode_tsit5_kernel(const float* __restrict__ x0, const float* __restrict__ u,
                 const float* __restrict__ t0p, const float* __restrict__ t1p,
                 const float* __restrict__ W0, const float* __restrict__ b0,
                 const float* __restrict__ W1, const float* __restrict__ b1,
                 const float* __restrict__ W2, const float* __restrict__ b2,
                 float* __restrict__ out)
{
    extern __shared__ float smem[];
    const int tid  = threadIdx.x;
    const int wave = tid >> 5;
    const int lane = tid & 31;
    const int row0 = blockIdx.x * TILE;

    float* __restrict__ z    = smem + OFF_Z;
    float* __restrict__ sy   = smem + OFF_Y;
    float* __restrict__ sy5  = smem + OFF_Y5;
    float* __restrict__ k1   = smem + OFF_K + 0 * TILE * STATE;
    float* __restrict__ k2   = smem + OFF_K + 1 * TILE * STATE;
    float* __restrict__ k3   = smem + OFF_K + 2 * TILE * STATE;
    float* __restrict__ k4   = smem + OFF_K + 3 * TILE * STATE;
    float* __restrict__ k5   = smem + OFF_K + 4 * TILE * STATE;
    float* __restrict__ k6   = smem + OFF_K + 5 * TILE * STATE;
    float* __restrict__ k7   = smem + OFF_K + 6 * TILE * STATE;
    float* __restrict__ st   = smem + OFF_T;
    float* __restrict__ sdt  = smem + OFF_DT;
    float* __restrict__ sdtc = smem + OFF_DTC;
    float* __restrict__ sstep = smem + OFF_STEP;
    float* __restrict__ sdone = smem + OFF_DONE;
    int*   __restrict__ snst  = (int*)(smem + OFF_NST);

    // ---- stage weights into LDS, transposed: Wt[k][n] = W[n][k] ----
    for (int i = tid; i < WIDTH * ZDIM; i += NTHREADS) {
        int n = i / ZDIM, k = i % ZDIM;
        smem[OFF_W0T + k * WIDTH + n] = W0[i];
    }
    for (int i = tid; i < WIDTH * WIDTH; i += NTHREADS) {
        int n = i >> 7, k = i & 127;
        smem[OFF_W1T + k * WIDTH + n] = W1[i];
    }
    for (int i = tid; i < STATE * WIDTH; i += NTHREADS) {
        int n = i >> 7, k = i & 127;
        smem[OFF_W2T + k * STATE + n] = W2[i];
    }
    for (int i = tid; i < WIDTH; i += NTHREADS) {
        smem[OFF_B0 + i] = b0[i];
        smem[OFF_B1 + i] = b1[i];
    }
    if (tid < STATE) smem[OFF_B2 + tid] = b2[tid];

    // controls (fixed over the whole solve) into z columns [64,80)
    for (int i = tid; i < TILE * CTRL; i += NTHREADS) {
        int m = i >> 4, c = i & 15;
        z[m * SZ_LD + STATE + c] = u[(row0 + m) * CTRL + c];
    }
    // initial state
    for (int i = tid; i < TILE * STATE; i += NTHREADS)
        sy[i] = x0[row0 * STATE + i];

    const float t1 = t1p[0];
    if (tid < TILE) {
        st[tid]   = t0p[0];
        sdt[tid]  = DT0f;
        snst[tid] = 0;
    }
    __syncthreads();

    // ======================= fixed-length adaptive scan =====================
    for (int step = 0; step < MAX_STEPS; ++step) {
        if (tid < TILE) {
            float t = st[tid];
            bool done = (t >= t1 - 1e-12f);
            sdone[tid] = done ? 1.0f : 0.0f;
            sdtc[tid]  = fminf(sdt[tid], t1 - t);
        }
        __syncthreads();

        // ---- k1 = f(y) ----
        for (int i = tid; i < TILE * STATE; i += NTHREADS) {
            int m = i >> 6, c = i & 63;
            z[m * SZ_LD + c] = sy[i];
        }
        __syncthreads();
        mlp_eval(smem, k1, wave, lane);

        // ---- k2 ----
        for (int i = tid; i < TILE * STATE; i += NTHREADS) {
            int m = i >> 6, c = i & 63;
            z[m * SZ_LD + c] = sy[i] + sdtc[m] * (A21f * k1[i]);
        }
        __syncthreads();
        mlp_eval(smem, k2, wave, lane);

        // ---- k3 ----
        for (int i = tid; i < TILE * STATE; i += NTHREADS) {
            int m = i >> 6, c = i & 63;
            z[m * SZ_LD + c] = sy[i] + sdtc[m] * (A31f * k1[i] + A32f * k2[i]);
        }
        __syncthreads();
        mlp_eval(smem, k3, wave, lane);

        // ---- k4 ----
        for (int i = tid; i < TILE * STATE; i += NTHREADS) {
            int m = i >> 6, c = i & 63;
            z[m * SZ_LD + c] =
                sy[i] + sdtc[m] * (A41f * k1[i] + A42f * k2[i] + A43f * k3[i]);
        }
        __syncthreads();
        mlp_eval(smem, k4, wave, lane);

        // ---- k5 ----
        for (int i = tid; i < TILE * STATE; i += NTHREADS) {
            int m = i >> 6, c = i & 63;
            z[m * SZ_LD + c] = sy[i] + sdtc[m] * (A51f * k1[i] + A52f * k2[i] +
                                                  A53f * k3[i] + A54f * k4[i]);
        }
        __syncthreads();
        mlp_eval(smem, k5, wave, lane);

        // ---- k6 ----
        for (int i = tid; i < TILE * STATE; i += NTHREADS) {
            int m = i >> 6, c = i & 63;
            z[m * SZ_LD + c] =
                sy[i] + sdtc[m] * (A61f * k1[i] + A62f * k2[i] + A63f * k3[i] +
                                   A64f * k4[i] + A65f * k5[i]);
        }
        __syncthreads();
        mlp_eval(smem, k6, wave, lane);

        // ---- y5 (5th-order solution) and k7 = f(y5) ----
        for (int i = tid; i < TILE * STATE; i += NTHREADS) {
            int m = i >> 6, c = i & 63;
            float acc = B1f * k1[i] + B2f * k2[i] + B3f * k3[i] +
                        B4f * k4[i] + B5f * k5[i] + B6f * k6[i];
            float v = sy[i] + sdtc[m] * acc;
            sy5[i] = v;
            z[m * SZ_LD + c] = v;
        }
        __syncthreads();
        mlp_eval(smem, k7, wave, lane);

        // ---- error norm + controller (one row per thread; cheap vs GEMMs) --
        if (tid < TILE) {
            const int m = tid;
            const float d = sdtc[m];
            float sum = 0.0f;
            for (int c = 0; c < STATE; ++c) {
                const int i = m * STATE + c;
                float err = d * (E1f * k1[i] + E2f * k2[i] + E3f * k3[i] +
                                 E4f * k4[i] + E5f * k5[i] + E6f * k6[i] +
                                 E7f * k7[i]);
                float sc = ATOLf + RTOLf * fmaxf(fabsf(sy[i]), fabsf(sy5[i]));
                float r = err / sc;
                sum += r * r;
            }
            float en  = sqrtf(sum * (1.0f / STATE));
            bool accept = (en <= 1.0f);
            float fac = SAFETYf * powf(fmaxf(en, 1e-16f), -EXPf);
            fac = fminf(fmaxf(fac, FMINf), FMAXf);
            bool done = (sdone[m] > 0.5f);
            bool stp  = accept && !done;
            if (stp) st[m] = st[m] + d;
            sstep[m] = stp ? 1.0f : 0.0f;
            if (!done) {
                sdt[m] = d * fac;
                snst[m] += 1;
            }
        }
        __syncthreads();

        // ---- masked state update ----
        for (int i = tid; i < TILE * STATE; i += NTHREADS) {
            int m = i >> 6;
            if (sstep[m] > 0.5f) sy[i] = sy5[i];
        }
        __syncthreads();
    }

    // ---- outputs: y (1024x64 f32) then nsteps (1024 i32), concatenated ----
    for (int i = tid; i < TILE * STATE; i += NTHREADS)
        out[row0 * STATE + i] = sy[i];
    if (tid < TILE) {
        int* nout = (int*)out + N_B * STATE;
        nout[row0 + tid] = snst[tid];
    }
}

// ---------------------------------------------------------------------------
extern "C" void kernel_launch(void* const* d_in, const int* in_sizes, int n_in,
                              void* d_out, int out_size, void* d_ws, size_t ws_size,
                              hipStream_t stream)
{
    const float* x0 = (const float*)d_in[0];
    const float* u  = (const float*)d_in[1];
    const float* t0 = (const float*)d_in[2];
    const float* t1 = (const float*)d_in[3];
    const float* W0 = (const float*)d_in[4];
    const float* b0 = (const float*)d_in[5];
    const float* W1 = (const float*)d_in[6];
    const float* b1 = (const float*)d_in[7];
    const float* W2 = (const float*)d_in[8];
    const float* b2 = (const float*)d_in[9];
    float* out = (float*)d_out;

    const size_t smem_bytes = (size_t)SMEM_FLOATS * sizeof(float);
    hipLaunchKernelGGL(ode_tsit5_kernel,
                       dim3(N_B / TILE), dim3(NTHREADS), smem_bytes, stream,
                       x0, u, t0, t1, W0, b0, W1, b1, W2, b2, out);
}